// HarrisNet_8770323218829
// MI455X (gfx1250) — compile-verified
//
#include <hip/hip_runtime.h>
#include <math.h>

typedef __attribute__((ext_vector_type(2))) float v2f;
typedef __attribute__((ext_vector_type(8))) float v8f;

#define IMG_H 2048
#define IMG_W 2048
#define NIMG 4
#define NCH 3
#define TS 32          // output tile
#define XT 40          // x tile incl. halo 4 (3 gauss + 1 sobel)
#define PRR 38         // prods rows/cols (halo 3)

// ---------------------------------------------------------------------------
// Kernel 1: fused Sobel -> products -> separable 7x7 Gaussian (WMMA) -> S
// ---------------------------------------------------------------------------
__global__ __launch_bounds__(128) void k_sobel_gauss(
    const float* __restrict__ x, const float* __restrict__ sobel_w,
    const float* __restrict__ gauss_w, float* __restrict__ S) {
  __shared__ float xs[XT][XT];          // 40x40 input tile (zero-padded)
  __shared__ float P[NCH][PRR][XT];     // 38 rows x 40 cols (cols 38..39 = 0)
  __shared__ float Hh[NCH][40][36];     // horizontal pass; rows 38..39 = 0
  __shared__ float sw[2][9];
  __shared__ float g1[8];

  const int tid  = threadIdx.x;
  const int lane = tid & 31;
  const int wave = tid >> 5;
  const int tx = blockIdx.x * TS;
  const int ty = blockIdx.y * TS;
  const int n  = blockIdx.z;
  const float* xim = x + (size_t)n * IMG_H * IMG_W;

  if (tid < 18) sw[tid / 9][tid % 9] = sobel_w[tid];
  if (tid < 7)  g1[tid] = sqrtf(gauss_w[tid * 7 + tid]);  // separable factor
  if (tid == 7) g1[7] = 0.0f;

  // ---- load x tile, zero outside image (zero padding of both convs) ----
  for (int i = tid; i < XT * XT; i += 128) {
    int r = i / XT, c = i % XT;
    int gr = ty - 4 + r, gc = tx - 4 + c;
    float v = 0.0f;
    if (gr >= 0 && gr < IMG_H && gc >= 0 && gc < IMG_W)
      v = xim[(size_t)gr * IMG_W + gc];
    xs[r][c] = v;
  }
  __syncthreads();

  // ---- Sobel + structure-tensor products on 38x40 (pixel-OOB -> 0) ----
  for (int i = tid; i < PRR * XT; i += 128) {
    int r = i / XT, c = i % XT;
    float pxx = 0.0f, pyy = 0.0f, pxy = 0.0f;
    int pr = ty - 3 + r, pc = tx - 3 + c;
    if (c < PRR && pr >= 0 && pr < IMG_H && pc >= 0 && pc < IMG_W) {
      float ix = 0.0f, iy = 0.0f;
#pragma unroll
      for (int dy = 0; dy < 3; ++dy)
#pragma unroll
        for (int dx = 0; dx < 3; ++dx) {
          float v = xs[r + dy][c + dx];
          ix += sw[0][dy * 3 + dx] * v;
          iy += sw[1][dy * 3 + dx] * v;
        }
      pxx = ix * ix; pyy = iy * iy; pxy = ix * iy;
    }
    P[0][r][c] = pxx; P[1][r][c] = pyy; P[2][r][c] = pxy;
  }
  __syncthreads();

  // ---- per-lane band fragment, shared by H-pass (B) and V-pass (A) ----
  // band[s][v] = g1[(4s + kb + v) - (lane&15)]   (banded Toeplitz of g1)
  const int m  = lane & 15;               // A row / B,C,D column
  const int kb = (lane >> 4) * 2;         // K sub-offset from lane half
  v2f band[6];
#pragma unroll
  for (int s = 0; s < 6; ++s)
#pragma unroll
    for (int v = 0; v < 2; ++v) {
      int d = 4 * s + kb + v - m;
      band[s][v] = g1[(unsigned)d < 7u ? d : 7];   // g1[7] == 0
    }

  // ---- horizontal Gaussian, rows 0..31 via WMMA f32 16x16x4 ----
  // Out[r0+mm][c0+nn] = sum_K P[r0+mm][c0+K] * g1[K-nn]
#pragma unroll
  for (int it = 0; it < 3; ++it) {        // ch(3) x r0{0,16} x c0{0,16} = 12
    int st = wave + it * 4;
    int ch = st >> 2;
    int r0 = ((st >> 1) & 1) * 16;
    int c0 = (st & 1) * 16;
    v8f acc = {};
#pragma unroll
    for (int s = 0; s < 6; ++s) {         // K = 24 (band zero-padded)
      v2f a;
      a[0] = P[ch][r0 + m][c0 + 4 * s + kb];
      a[1] = P[ch][r0 + m][c0 + 4 * s + kb + 1];
      acc = __builtin_amdgcn_wmma_f32_16x16x4_f32(false, a, false, band[s],
                                                  (short)0, acc, false, false);
    }
#pragma unroll
    for (int v = 0; v < 8; ++v)
      Hh[ch][r0 + v + (lane >> 4) * 8][c0 + m] = acc[v];
  }
  // rows 32..37 direct; rows 38..39 zero (K-padding safety)
  for (int i = tid; i < NCH * 8 * TS; i += 128) {
    int ch = i / (8 * TS);
    int rem = i % (8 * TS);
    int r = 32 + rem / TS;
    int c = rem % TS;
    float v = 0.0f;
    if (r < PRR) {
#pragma unroll
      for (int k = 0; k < 7; ++k) v += g1[k] * P[ch][r][c + k];
    }
    Hh[ch][r][c] = v;
  }
  __syncthreads();

  // ---- vertical Gaussian via WMMA, store S ----
  // Out[r0+mm][c0+nn] = sum_q g1[q-mm] * Hh[r0+q][c0+nn]
#pragma unroll
  for (int it = 0; it < 3; ++it) {
    int st = wave + it * 4;
    int ch = st >> 2;
    int r0 = ((st >> 1) & 1) * 16;
    int c0 = (st & 1) * 16;
    v8f acc = {};
#pragma unroll
    for (int s = 0; s < 6; ++s) {
      v2f b;
      b[0] = Hh[ch][r0 + 4 * s + kb][c0 + m];
      b[1] = Hh[ch][r0 + 4 * s + kb + 1][c0 + m];
      acc = __builtin_amdgcn_wmma_f32_16x16x4_f32(false, band[s], false, b,
                                                  (short)0, acc, false, false);
    }
#pragma unroll
    for (int v = 0; v < 8; ++v) {
      int row = ty + r0 + v + (lane >> 4) * 8;
      int col = tx + c0 + m;
      S[((size_t)(n * NCH + ch) * IMG_H + row) * IMG_W + col] = acc[v];
    }
  }
}

// ---------------------------------------------------------------------------
// Exact global median via 2-pass 16-bit radix histogram
// ---------------------------------------------------------------------------
__device__ __forceinline__ unsigned fkey(float f) {
  unsigned u = __float_as_uint(f);
  return (u & 0x80000000u) ? ~u : (u | 0x80000000u);  // order-preserving
}

__global__ void k_zero(unsigned* __restrict__ p, int n) {
  int i = blockIdx.x * blockDim.x + threadIdx.x;
  if (i < n) p[i] = 0u;
}

__global__ void k_hist1(const float* __restrict__ S, unsigned* __restrict__ hist,
                        size_t nelem) {
  size_t stride = (size_t)gridDim.x * blockDim.x;
  for (size_t i = (size_t)blockIdx.x * blockDim.x + threadIdx.x; i < nelem;
       i += stride) {
    if (i + stride < nelem) __builtin_prefetch(&S[i + stride], 0, 1);
    atomicAdd(&hist[fkey(S[i]) >> 16], 1u);
  }
}

__global__ void k_select1(const unsigned* __restrict__ hist,
                          unsigned* __restrict__ sel, unsigned long long k) {
  if (threadIdx.x == 0 && blockIdx.x == 0) {
    unsigned long long c = 0;
    for (int b = 0; b < 65536; ++b) {
      unsigned long long nc = c + hist[b];
      if (nc > k) { sel[0] = (unsigned)b; sel[1] = (unsigned)(k - c); break; }
      c = nc;
    }
  }
}

__global__ void k_hist2(const float* __restrict__ S,
                        const unsigned* __restrict__ sel,
                        unsigned* __restrict__ hist, size_t nelem) {
  unsigned B = sel[0];
  size_t stride = (size_t)gridDim.x * blockDim.x;
  for (size_t i = (size_t)blockIdx.x * blockDim.x + threadIdx.x; i < nelem;
       i += stride) {
    if (i + stride < nelem) __builtin_prefetch(&S[i + stride], 0, 1);
    unsigned key = fkey(S[i]);
    if ((key >> 16) == B) atomicAdd(&hist[key & 0xFFFFu], 1u);
  }
}

__global__ void k_select2(const unsigned* __restrict__ hist,
                          const unsigned* __restrict__ sel,
                          float* __restrict__ med) {
  if (threadIdx.x == 0 && blockIdx.x == 0) {
    unsigned long long k = sel[1], c = 0;
    unsigned low = 0;
    for (int b = 0; b < 65536; ++b) {
      unsigned long long nc = c + hist[b];
      if (nc > k) { low = (unsigned)b; break; }
      c = nc;
    }
    unsigned key = (sel[0] << 16) | low;
    unsigned u = (key & 0x80000000u) ? (key & 0x7FFFFFFFu) : ~key;
    *med = __uint_as_float(u);
  }
}

// ---------------------------------------------------------------------------
// Kernel: threshold + separable 7x7 max-pool NMS
// ---------------------------------------------------------------------------
__global__ __launch_bounds__(256) void k_nms(const float* __restrict__ S,
                                             const float* __restrict__ medp,
                                             float* __restrict__ out) {
  __shared__ float xt[PRR][40];
  __shared__ float hm[PRR][36];
  const int tid = threadIdx.x;
  const float med = *medp;
  const int tx = blockIdx.x * TS, ty = blockIdx.y * TS;
  const int nc = blockIdx.z;
  const float* Sp = S + (size_t)nc * IMG_H * IMG_W;

  for (int i = tid; i < PRR * PRR; i += 256) {
    int r = i / PRR, c = i % PRR;
    int gr = ty - 3 + r, gc = tx - 3 + c;
    float v;
    if (gr >= 0 && gr < IMG_H && gc >= 0 && gc < IMG_W) {
      float s = Sp[(size_t)gr * IMG_W + gc];
      v = (s > med) ? s : 0.0f;
    } else {
      v = -__builtin_inff();   // reduce_window pad value
    }
    xt[r][c] = v;
  }
  __syncthreads();
  for (int i = tid; i < PRR * TS; i += 256) {   // horizontal running max
    int r = i / TS, c = i % TS;
    float mx = xt[r][c];
#pragma unroll
    for (int k = 1; k < 7; ++k) mx = fmaxf(mx, xt[r][c + k]);
    hm[r][c] = mx;
  }
  __syncthreads();
  for (int i = tid; i < TS * TS; i += 256) {    // vertical max + NMS
    int r = i / TS, c = i % TS;
    float y = hm[r][c];
#pragma unroll
    for (int k = 1; k < 7; ++k) y = fmaxf(y, hm[r + k][c]);
    float v = xt[r + 3][c + 3];
    out[((size_t)nc * IMG_H + ty + r) * IMG_W + tx + c] = (v == y) ? v : 0.0f;
  }
}

// ---------------------------------------------------------------------------
extern "C" void kernel_launch(void* const* d_in, const int* in_sizes, int n_in,
                              void* d_out, int out_size, void* d_ws,
                              size_t ws_size, hipStream_t stream) {
  const float* x       = (const float*)d_in[0];
  const float* sobel_w = (const float*)d_in[1];
  const float* gauss_w = (const float*)d_in[2];
  float* out = (float*)d_out;

  const size_t NS = (size_t)NIMG * NCH * IMG_H * IMG_W;  // 50,331,648
  float*    S     = (float*)d_ws;
  unsigned* hist1 = (unsigned*)(S + NS);
  unsigned* hist2 = hist1 + 65536;
  unsigned* sel   = hist2 + 65536;
  float*    med   = (float*)(sel + 4);

  dim3 g1(IMG_W / TS, IMG_H / TS, NIMG);
  k_sobel_gauss<<<g1, 128, 0, stream>>>(x, sobel_w, gauss_w, S);

  int nz = 65536 * 2 + 8;
  k_zero<<<(nz + 255) / 256, 256, 0, stream>>>(hist1, nz);

  k_hist1<<<2048, 256, 0, stream>>>(S, hist1, NS);
  const unsigned long long k = (NS - 1) / 2;  // quantile 0.5, method='lower'
  k_select1<<<1, 32, 0, stream>>>(hist1, sel, k);
  k_hist2<<<2048, 256, 0, stream>>>(S, sel, hist2, NS);
  k_select2<<<1, 32, 0, stream>>>(hist2, sel, med);

  dim3 g2(IMG_W / TS, IMG_H / TS, NIMG * NCH);
  k_nms<<<g2, 256, 0, stream>>>(S, med, out);
}